// DNCMemory_68822555951214
// MI455X (gfx1250) — compile-verified
//
#include <hip/hip_runtime.h>
#include <math.h>

// Problem constants from the reference: B=32, N=1024, W=64, R=4, NW=1
#define BB 32
#define NN 1024
#define WWID 64
#define RR 4
#define EPSF 1e-5f

typedef __attribute__((ext_vector_type(2))) float v2f;
typedef __attribute__((ext_vector_type(8))) float v8f;
typedef __attribute__((ext_vector_type(4))) unsigned int u32x4;
typedef __attribute__((ext_vector_type(8))) unsigned int u32x8;

// ---------------------------------------------------------------------------
// TDM: DMA a 1024x16 f32 panel (row-major, stride 16) from global into LDS.
// D# per CDNA5 ISA:
//   group0: [1:0]=count=1, [63:32]=lds_addr, [120:64]=global_addr,
//           [127:126]=type=2
//   group1: [17:16]=data_size=2 (4B), tensor_dim0=16, tensor_dim1=1024,
//           tile_dim0=16, tile_dim1=1024, tensor_dim0_stride=16
// Groups 2/3 are NULL (2-D tensor). Issued by one wave (TDM ignores EXEC,
// one issue per wave); completion via TENSORcnt.
// Note: low 32 bits of a flat LDS pointer are the LDS byte offset (the
// hardware maps flat LDS addresses by truncating to addr[31:0]).
// ---------------------------------------------------------------------------
__device__ __forceinline__ void tdm_load_panel(const float* gsrc,
                                               void* lds_dst) {
  unsigned long long ga = (unsigned long long)(const void*)gsrc;
  unsigned lds_off = (unsigned)(unsigned long long)lds_dst;
  u32x4 g0;
  g0.x = 1u;                                                  // count=1
  g0.y = lds_off;                                             // lds_addr
  g0.z = (unsigned)ga;                                        // addr[31:0]
  g0.w = (unsigned)((ga >> 32) & 0x1FFFFFFull) | (2u << 30);  // addr[56:32]|type=2
  u32x8 g1;
  g1[0] = (2u << 16);     // workgroup_mask=0, data_size=2 (4 bytes)
  g1[1] = (16u << 16);    // tensor_dim0 = 16 (bits 63:48)
  g1[2] = (1024u << 16);  // tensor_dim1 = 1024 (bits 95:80)
  g1[3] = (16u << 16);    // tile_dim0 = 16 (bits 127:112)
  g1[4] = 1024u;          // tile_dim1 = 1024 (bits 143:128)
  g1[5] = 16u;            // tensor_dim0_stride = 16 (bits 191:160)
  g1[6] = 0u;
  g1[7] = 0u;
  asm volatile("tensor_load_to_lds %0, %1" ::"s"(g0), "s"(g1) : "memory");
  __builtin_amdgcn_s_wait_tensorcnt(0);
}

// ---------------------------------------------------------------------------
// 1) usage = (u + (1-u)*w_agg) * prod_r(1 - fg_r * rwp_r)     (NW==1)
// ---------------------------------------------------------------------------
__global__ void usage_kernel(const float* __restrict__ usage_prev,
                             const float* __restrict__ wwp,
                             const float* __restrict__ free_gate,
                             const float* __restrict__ rwp,
                             float* __restrict__ usage) {
  int b = blockIdx.x;
  for (int n = threadIdx.x; n < NN; n += blockDim.x) {
    float u = usage_prev[b * NN + n];
    float wa = wwp[b * NN + n];  // NW==1: w_agg = ww
    u = u + (1.f - u) * wa;
    float phi = 1.f;
#pragma unroll
    for (int r = 0; r < RR; ++r)
      phi *= 1.f - free_gate[b * RR + r] * rwp[(b * RR + r) * NN + n];
    usage[b * NN + n] = u * phi;
  }
}

// ---------------------------------------------------------------------------
// 2) allocation weighting: bitonic sort (desc by nonusage, stable) in LDS,
//    prefix-product scan, scatter back through sort indices.
// ---------------------------------------------------------------------------
__global__ void alloc_kernel(const float* __restrict__ usage,
                             float* __restrict__ alloc) {
  __shared__ float vals[NN];
  __shared__ int sidx[NN];
  __shared__ float sa[NN];
  int b = blockIdx.x, tid = threadIdx.x;
  vals[tid] = 1.f - usage[b * NN + tid];  // nonusage
  sidx[tid] = tid;
  __syncthreads();
  for (int k = 2; k <= NN; k <<= 1) {
    for (int j = k >> 1; j > 0; j >>= 1) {
      int i = tid, ixj = i ^ j;
      if (ixj > i) {
        float a = vals[i], c = vals[ixj];
        int ia = sidx[i], ic = sidx[ixj];
        bool desc = ((i & k) == 0);
        bool a_first = (a > c) || (a == c && ia < ic);  // stable top_k order
        if (desc ? !a_first : a_first) {
          vals[i] = c; vals[ixj] = a;
          sidx[i] = ic; sidx[ixj] = ia;
        }
      }
      __syncthreads();
    }
  }
  float nu = vals[tid];
  sa[tid] = 1.f - nu;  // sorted usage
  __syncthreads();
  for (int off = 1; off < NN; off <<= 1) {  // inclusive prefix product
    float t = (tid >= off) ? sa[tid - off] : 1.f;
    __syncthreads();
    sa[tid] *= t;
    __syncthreads();
  }
  float excl = (tid == 0) ? 1.f : sa[tid - 1];
  alloc[b * NN + sidx[tid]] = nu * excl;
}

// ---------------------------------------------------------------------------
// 3) cosine-similarity content weights + softmax over N. One block per (b,h).
// ---------------------------------------------------------------------------
__global__ void content_kernel(const float* __restrict__ mem,
                               const float* __restrict__ keys,
                               const float* __restrict__ strengths,
                               float* __restrict__ out, int H) {
  int b = blockIdx.x / H, h = blockIdx.x % H;
  __shared__ float key[WWID];
  __shared__ float sharp[NN];
  __shared__ float red[256];
  int tid = threadIdx.x;
  if (tid < WWID) key[tid] = keys[(b * H + h) * WWID + tid];
  __syncthreads();
  float kn2 = 0.f;
#pragma unroll
  for (int w = 0; w < WWID; ++w) kn2 += key[w] * key[w];
  float knorm = sqrtf(kn2 + EPSF);
  float s = strengths[b * H + h];
  float sp = (s > 20.f) ? s : log1pf(expf(s));  // softplus
  for (int n = tid; n < NN; n += 256) {
    const float* m = mem + ((size_t)(b * NN + n)) * WWID;
    float dot = 0.f, mn2 = 0.f;
#pragma unroll
    for (int w = 0; w < WWID; ++w) {
      float mv = m[w];
      dot += key[w] * mv;
      mn2 += mv * mv;
    }
    float mnorm = sqrtf(mn2 + EPSF);
    sharp[n] = dot / (knorm * mnorm + EPSF) * sp;
  }
  __syncthreads();
  float mx = -INFINITY;
  for (int n = tid; n < NN; n += 256) mx = fmaxf(mx, sharp[n]);
  red[tid] = mx;
  __syncthreads();
  for (int o = 128; o > 0; o >>= 1) {
    if (tid < o) red[tid] = fmaxf(red[tid], red[tid + o]);
    __syncthreads();
  }
  mx = red[0];
  __syncthreads();
  float sum = 0.f;
  for (int n = tid; n < NN; n += 256) sum += expf(sharp[n] - mx);
  red[tid] = sum;
  __syncthreads();
  for (int o = 128; o > 0; o >>= 1) {
    if (tid < o) red[tid] += red[tid + o];
    __syncthreads();
  }
  sum = red[0];
  for (int n = tid; n < NN; n += 256)
    out[(b * H + h) * NN + n] = expf(sharp[n] - mx) / sum;
}

// ---------------------------------------------------------------------------
// 4) write_weights = wg * (ag*alloc + (1-ag)*write_content)       (NW==1)
// ---------------------------------------------------------------------------
__global__ void ww_kernel(const float* __restrict__ alloc,
                          const float* __restrict__ wcontent,
                          const float* __restrict__ agate,
                          const float* __restrict__ wgate,
                          float* __restrict__ ww) {
  int b = blockIdx.x;
  float ag = agate[b], wg = wgate[b];
  for (int n = threadIdx.x; n < NN; n += blockDim.x)
    ww[b * NN + n] =
        wg * (ag * alloc[b * NN + n] + (1.f - ag) * wcontent[b * NN + n]);
}

// ---------------------------------------------------------------------------
// 5) memory_new = memory*(1 - w[n]*erase[c]) + w[n]*write_vec[c]   (NW==1)
// ---------------------------------------------------------------------------
__global__ void memup_kernel(const float* __restrict__ mem,
                             const float* __restrict__ ww,
                             const float* __restrict__ erase,
                             const float* __restrict__ wvec,
                             float* __restrict__ mem_new) {
  size_t i = (size_t)blockIdx.x * blockDim.x + threadIdx.x;
  if (i >= (size_t)BB * NN * WWID) return;
  int c = (int)(i % WWID);
  size_t bn = i / WWID;
  int b = (int)(bn / NN);
  float w = ww[bn];
  mem_new[i] = mem[i] * (1.f - w * erase[b * WWID + c]) + w * wvec[b * WWID + c];
}

// ---------------------------------------------------------------------------
// 6) Pack the 16-column operand panels for the link matvecs (cols 8..15 = 0)
//    X cols: [a_0..a_3, a_0*w..a_3*w],  Y cols: [a_r*(1-w), a_r]
// ---------------------------------------------------------------------------
__global__ void prep_xy_kernel(const float* __restrict__ rwp,
                               const float* __restrict__ ww,
                               float* __restrict__ X, float* __restrict__ Y) {
  int b = blockIdx.x;
  for (int n = threadIdx.x; n < NN; n += blockDim.x) {
    float w = ww[b * NN + n];
    float* x = X + ((size_t)b * NN + n) * 16;
    float* y = Y + ((size_t)b * NN + n) * 16;
#pragma unroll
    for (int r = 0; r < RR; ++r) {
      float a = rwp[(b * RR + r) * NN + n];
      x[r] = a;
      x[4 + r] = a * w;
      y[r] = a * (1.f - w);
      y[4 + r] = a;
    }
#pragma unroll
    for (int j = 8; j < 16; ++j) {
      x[j] = 0.f;
      y[j] = 0.f;
    }
  }
}

// rank-1 correction scalars: S1[b,r] = a_r . p,  S2[b,r] = a_r . w
__global__ void dots_kernel(const float* __restrict__ rwp,
                            const float* __restrict__ prec,
                            const float* __restrict__ ww,
                            float* __restrict__ S1, float* __restrict__ S2) {
  int br = blockIdx.x;
  int b = br / RR;
  __shared__ float r1[256], r2[256];
  int tid = threadIdx.x;
  float s1 = 0.f, s2 = 0.f;
  for (int n = tid; n < NN; n += 256) {
    float a = rwp[(size_t)br * NN + n];
    s1 += a * prec[b * NN + n];
    s2 += a * ww[b * NN + n];
  }
  r1[tid] = s1;
  r2[tid] = s2;
  __syncthreads();
  for (int o = 128; o > 0; o >>= 1) {
    if (tid < o) {
      r1[tid] += r1[tid + o];
      r2[tid] += r2[tid + o];
    }
    __syncthreads();
  }
  if (tid == 0) {
    S1[br] = r1[0];
    S2[br] = r2[0];
  }
}

// ---------------------------------------------------------------------------
// 7) fwd_w via WMMA. 8 waves per block; wave w owns row tile blockIdx.x*8+w.
//    X panel (1024x16, 64KB) is TDM-staged into LDS once per block; WMMA
//    B-fragments then come from LDS (ds_load), A streams link rows from HBM.
//    A frag (16x4 f32): lanes 0-15 hold K={0,1}, lanes 16-31 hold K={2,3}
//    -> one contiguous float2 per lane per chunk.
//    After the K loop the LDS panel is reused (post-barrier) as per-wave
//    16x16 D-tile scratch for the cross-lane recombination.
// ---------------------------------------------------------------------------
__global__ void __launch_bounds__(256) fwd_wmma_kernel(
    const float* __restrict__ link, const float* __restrict__ X,
    const float* __restrict__ ww, const float* __restrict__ S1,
    float* __restrict__ fwd) {
  __shared__ float Xs[NN * 16];  // 64KB panel, aliased later as D-tile scratch
  int b = blockIdx.y;
  int wave = threadIdx.x >> 5;
  int lane = threadIdx.x & 31;
  int n0 = (blockIdx.x * 8 + wave) * 16;
  int row = lane & 15;
  int kbase = (lane < 16) ? 0 : 2;
  const float* Xb = X + (size_t)b * NN * 16;
  if (wave == 0) tdm_load_panel(Xb, (void*)Xs);
  __syncthreads();
  const float* Lrow = link + (size_t)b * NN * NN + (size_t)(n0 + row) * NN;
  v8f acc = {};
  for (int m = 0; m < NN; m += 4) {
    if ((m & 127) == 0)  // uniform branch: stream-ahead prefetch of link row
      __builtin_prefetch(Lrow + m + 128, 0, 1);
    v2f a;
    a.x = Lrow[m + kbase];
    a.y = Lrow[m + kbase + 1];
    v2f bv;
    bv.x = Xs[(m + kbase) * 16 + row];
    bv.y = Xs[(m + kbase + 1) * 16 + row];
    acc = __builtin_amdgcn_wmma_f32_16x16x4_f32(false, a, false, bv, (short)0,
                                                acc, false, false);
  }
  __syncthreads();  // all waves done reading the panel -> reuse LDS
  float* tile = Xs + wave * 256;  // D: row = v + 8*(lane>=16), col = lane&15
  int mrow = (lane < 16) ? 0 : 8;
#pragma unroll
  for (int v = 0; v < 8; ++v) tile[(mrow + v) * 16 + row] = acc[v];
  __syncthreads();
  for (int t = lane; t < 64; t += 32) {
    int rr = t & 3, rw = t >> 2;
    int n = n0 + rw;
    float w = ww[b * NN + n];
    fwd[((size_t)b * RR + rr) * NN + n] =
        (1.f - w) * tile[rw * 16 + rr] - tile[rw * 16 + 4 + rr] +
        w * S1[b * RR + rr];
  }
}

// ---------------------------------------------------------------------------
// 8) bwd_w via WMMA: A frag holds L transposed (contiguous 64B row segments),
//    B panel = Y (TDM-staged to LDS).
// ---------------------------------------------------------------------------
__global__ void __launch_bounds__(256) bwd_wmma_kernel(
    const float* __restrict__ link, const float* __restrict__ Y,
    const float* __restrict__ ww, const float* __restrict__ prec,
    const float* __restrict__ S2, float* __restrict__ bwd) {
  __shared__ float Ys[NN * 16];  // 64KB panel, aliased later as D-tile scratch
  int b = blockIdx.y;
  int wave = threadIdx.x >> 5;
  int lane = threadIdx.x & 31;
  int m0 = (blockIdx.x * 8 + wave) * 16;
  int row = lane & 15;  // output index within tile (L column)
  int kbase = (lane < 16) ? 0 : 2;
  const float* Yb = Y + (size_t)b * NN * 16;
  if (wave == 0) tdm_load_panel(Yb, (void*)Ys);
  __syncthreads();
  const float* Lb = link + (size_t)b * NN * NN;
  v8f acc = {};
  for (int n = 0; n < NN; n += 4) {
    if ((n & 127) == 0)
      __builtin_prefetch(Lb + (size_t)(n + 128) * NN + m0 + row, 0, 1);
    v2f a;
    a.x = Lb[(size_t)(n + kbase) * NN + m0 + row];
    a.y = Lb[(size_t)(n + kbase + 1) * NN + m0 + row];
    v2f bv;
    bv.x = Ys[(n + kbase) * 16 + row];
    bv.y = Ys[(n + kbase + 1) * 16 + row];
    acc = __builtin_amdgcn_wmma_f32_16x16x4_f32(false, a, false, bv, (short)0,
                                                acc, false, false);
  }
  __syncthreads();
  float* tile = Ys + wave * 256;
  int mrow = (lane < 16) ? 0 : 8;
#pragma unroll
  for (int v = 0; v < 8; ++v) tile[(mrow + v) * 16 + row] = acc[v];
  __syncthreads();
  for (int t = lane; t < 64; t += 32) {
    int rr = t & 3, q = t >> 2;
    int m = m0 + q;
    float w = ww[b * NN + m];
    bwd[((size_t)b * RR + rr) * NN + m] =
        tile[q * 16 + rr] - w * tile[q * 16 + 4 + rr] +
        prec[b * NN + m] * S2[b * RR + rr];
  }
}

// ---------------------------------------------------------------------------
// 9) read_weights = softmax(read_mode) . [bwd, fwd, content]      (NW==1)
// ---------------------------------------------------------------------------
__global__ void rw_kernel(const float* __restrict__ rcontent,
                          const float* __restrict__ fwd,
                          const float* __restrict__ bwd,
                          const float* __restrict__ read_mode,
                          float* __restrict__ rw) {
  int br = blockIdx.x;
  float m0 = read_mode[br * 3 + 0], m1 = read_mode[br * 3 + 1],
        m2 = read_mode[br * 3 + 2];
  float mx = fmaxf(m0, fmaxf(m1, m2));
  float e0 = expf(m0 - mx), e1 = expf(m1 - mx), e2 = expf(m2 - mx);
  float inv = 1.f / (e0 + e1 + e2);
  float bm = e0 * inv, fm = e1 * inv, cm = e2 * inv;
  for (int n = threadIdx.x; n < NN; n += blockDim.x) {
    size_t i = (size_t)br * NN + n;
    rw[i] = cm * rcontent[i] + fm * fwd[i] + bm * bwd[i];
  }
}

// ---------------------------------------------------------------------------
// 10) read_words[b,r,c] = sum_n rw[b,r,n] * mem_new[b,n,c]
// ---------------------------------------------------------------------------
__global__ void readwords_kernel(const float* __restrict__ rw,
                                 const float* __restrict__ mem_new,
                                 float* __restrict__ out) {
  int br = blockIdx.x;
  int b = br / RR;
  int c = threadIdx.x;  // 64 threads
  float acc = 0.f;
  for (int n = 0; n < NN; ++n)
    acc += rw[(size_t)br * NN + n] * mem_new[((size_t)b * NN + n) * WWID + c];
  out[br * WWID + c] = acc;
}

// ---------------------------------------------------------------------------
extern "C" void kernel_launch(void* const* d_in, const int* in_sizes, int n_in,
                              void* d_out, int out_size, void* d_ws,
                              size_t ws_size, hipStream_t stream) {
  const float* memory = (const float*)d_in[0];
  const float* usage_prev = (const float*)d_in[1];
  const float* link_prev = (const float*)d_in[2];
  const float* precedence_prev = (const float*)d_in[3];
  const float* read_weights_prev = (const float*)d_in[4];
  const float* write_weights_prev = (const float*)d_in[5];
  const float* read_keys = (const float*)d_in[6];
  const float* read_strengths = (const float*)d_in[7];
  const float* write_keys = (const float*)d_in[8];
  const float* write_strengths = (const float*)d_in[9];
  const float* erase_vectors = (const float*)d_in[10];
  const float* write_vectors = (const float*)d_in[11];
  const float* free_gate = (const float*)d_in[12];
  const float* allocation_gate = (const float*)d_in[13];
  const float* write_gate = (const float*)d_in[14];
  const float* read_mode = (const float*)d_in[15];
  float* out = (float*)d_out;

  float* ws = (float*)d_ws;
  float* usage = ws;                    // B*N
  float* alloc = usage + BB * NN;       // B*N
  float* wcont = alloc + BB * NN;       // B*N
  float* wwn = wcont + BB * NN;         // B*N
  float* memnew = wwn + BB * NN;        // B*N*W
  float* X = memnew + (size_t)BB * NN * WWID;  // B*N*16
  float* Y = X + (size_t)BB * NN * 16;         // B*N*16
  float* S1 = Y + (size_t)BB * NN * 16;        // B*R
  float* S2 = S1 + BB * RR;                    // B*R
  float* fwd = S2 + BB * RR;                   // B*R*N
  float* bwd = fwd + (size_t)BB * RR * NN;     // B*R*N
  float* rcont = bwd + (size_t)BB * RR * NN;   // B*R*N
  float* rwts = rcont + (size_t)BB * RR * NN;  // B*R*N

  usage_kernel<<<BB, 256, 0, stream>>>(usage_prev, write_weights_prev,
                                       free_gate, read_weights_prev, usage);
  alloc_kernel<<<BB, NN, 0, stream>>>(usage, alloc);
  content_kernel<<<BB, 256, 0, stream>>>(memory, write_keys, write_strengths,
                                         wcont, 1);
  ww_kernel<<<BB, 256, 0, stream>>>(alloc, wcont, allocation_gate, write_gate,
                                    wwn);
  {
    size_t tot = (size_t)BB * NN * WWID;
    memup_kernel<<<(unsigned)((tot + 255) / 256), 256, 0, stream>>>(
        memory, wwn, erase_vectors, write_vectors, memnew);
  }
  content_kernel<<<BB * RR, 256, 0, stream>>>(memnew, read_keys,
                                              read_strengths, rcont, RR);
  prep_xy_kernel<<<BB, 256, 0, stream>>>(read_weights_prev, wwn, X, Y);
  dots_kernel<<<BB * RR, 256, 0, stream>>>(read_weights_prev, precedence_prev,
                                           wwn, S1, S2);
  fwd_wmma_kernel<<<dim3(NN / 16 / 8, BB), 256, 0, stream>>>(link_prev, X, wwn,
                                                             S1, fwd);
  bwd_wmma_kernel<<<dim3(NN / 16 / 8, BB), 256, 0, stream>>>(
      link_prev, Y, wwn, precedence_prev, S2, bwd);
  rw_kernel<<<BB * RR, 256, 0, stream>>>(rcont, fwd, bwd, read_mode, rwts);
  readwords_kernel<<<BB * RR, WWID, 0, stream>>>(rwts, memnew, out);
  (void)in_sizes;
  (void)n_in;
  (void)out_size;
  (void)ws_size;
}